// AttentionPreferenceMiner_19997367730720
// MI455X (gfx1250) — compile-verified
//
#include <hip/hip_runtime.h>
#include <hip/hip_bf16.h>

typedef __attribute__((ext_vector_type(2))) float v2f;
typedef __attribute__((ext_vector_type(8))) float v8f;

#define EMB 64

// ---------------------------------------------------------------------------
// Monotonic float <-> uint key (total order incl. negatives) for atomicMax
// ---------------------------------------------------------------------------
__device__ __forceinline__ unsigned f2key(float f) {
    unsigned b = __float_as_uint(f);
    return (b & 0x80000000u) ? ~b : (b | 0x80000000u);
}
__device__ __forceinline__ float key2f(unsigned k) {
    unsigned b = (k & 0x80000000u) ? (k & 0x7FFFFFFFu) : ~k;
    return __uint_as_float(b);
}

// ---------------------------------------------------------------------------
// Kernel 1: Out[m,n] = sum_k X[m,k] * W[n,k]   (x @ W.T), exact fp32 via
// V_WMMA_F32_16X16X4_F32. One wave per 16x16 output tile, 16 K-steps.
// A 16x4 f32 layout: lane -> M = lane%16, K = (lane/16)*2 + j (j = VGPR idx)
// B 4x16 f32 layout: lane -> N = lane%16, K = (lane/16)*2 + j
// C/D 16x16 f32:     lane -> N = lane%16, M = (lane/16)*8 + j
// ---------------------------------------------------------------------------
__global__ void proj_wmma_f32(const float* __restrict__ X,
                              const float* __restrict__ W,
                              float* __restrict__ Out, int M) {
    const int wave = blockIdx.x * (blockDim.x >> 5) + (threadIdx.x >> 5);
    const int lane = threadIdx.x & 31;
    const int tileN = wave & 3;        // 64 / 16 = 4 N-tiles
    const int tileM = wave >> 2;
    if (tileM * 16 + 16 > M) return;   // uniform per wave -> EXEC stays all-1s

    const int mrow  = tileM * 16 + (lane & 15);
    const int ncol  = tileN * 16 + (lane & 15);
    const int khalf = (lane >> 4) * 2;

    const float* xa = X + (size_t)mrow * EMB + khalf;
    const float* wb = W + (size_t)ncol * EMB + khalf;

    v8f c = {};
#pragma unroll
    for (int k = 0; k < EMB; k += 4) {
        v2f a = *(const v2f*)(xa + k);
        v2f b = *(const v2f*)(wb + k);
        c = __builtin_amdgcn_wmma_f32_16x16x4_f32(
            /*neg_a=*/false, a, /*neg_b=*/false, b,
            /*c_mod=*/(short)0, c, /*reuse_a=*/false, /*reuse_b=*/false);
    }

    const int mbase = tileM * 16 + (lane >> 4) * 8;
    const int nout  = tileN * 16 + (lane & 15);
#pragma unroll
    for (int j = 0; j < 8; ++j)
        Out[(size_t)(mbase + j) * EMB + nout] = c[j];
}

// ---------------------------------------------------------------------------
// Kernel 2: zero init (segmax key = 0 == "below -inf", sums/cnt = 0, out = 0)
// ---------------------------------------------------------------------------
__global__ void init_zero(float* __restrict__ out, float* __restrict__ segsum,
                          float* __restrict__ cnt, unsigned* __restrict__ segmax,
                          int NU) {
    const int t = blockIdx.x * blockDim.x + threadIdx.x;
    if (t < NU * EMB) out[t] = 0.0f;
    if (t < NU) { segsum[t] = 0.0f; cnt[t] = 0.0f; segmax[t] = 0u; }
}

// ---------------------------------------------------------------------------
// Kernel 3: per-edge score + segment max
// ---------------------------------------------------------------------------
__global__ void edge_scores(const int* __restrict__ users,
                            const int* __restrict__ items,
                            const float* __restrict__ ew,
                            const float* __restrict__ Q,
                            const float* __restrict__ Kp,
                            float* __restrict__ scores,
                            unsigned* __restrict__ segmax, int E) {
    const int e = blockIdx.x * blockDim.x + threadIdx.x;
    if (e >= E) return;
    const int u = users[e];
    const int it = items[e];
    const float4* q = (const float4*)(Q + (size_t)u * EMB);
    const float4* k = (const float4*)(Kp + (size_t)it * EMB);
    float s = 0.0f;
#pragma unroll
    for (int j = 0; j < EMB / 4; ++j) {
        float4 a = q[j], b = k[j];
        s += a.x * b.x + a.y * b.y + a.z * b.z + a.w * b.w;
    }
    s *= 0.125f;          // 1/sqrt(64)
    s *= ew[e];
    scores[e] = s;
    atomicMax(segmax + u, f2key(s));
}

// ---------------------------------------------------------------------------
// Kernel 4: ex = exp(s - segmax[u]); segment sum + count
// ---------------------------------------------------------------------------
__global__ void edge_exp(const int* __restrict__ users,
                         const float* __restrict__ scores,
                         const unsigned* __restrict__ segmax,
                         float* __restrict__ ex,
                         float* __restrict__ segsum,
                         float* __restrict__ cnt, int E) {
    const int e = blockIdx.x * blockDim.x + threadIdx.x;
    if (e >= E) return;
    const int u = users[e];
    const float v = expf(scores[e] - key2f(segmax[u]));
    ex[e] = v;
    atomicAdd(segsum + u, v);
    atomicAdd(cnt + u, 1.0f);
}

// ---------------------------------------------------------------------------
// Kernel 5: scatter alpha * value into per-user sums (d_out), one thread per
// (edge, 4-dim chunk) -> 16-way parallel atomics per edge
// ---------------------------------------------------------------------------
__global__ void edge_scatter(const int* __restrict__ users,
                             const int* __restrict__ items,
                             const float* __restrict__ ex,
                             const float* __restrict__ segsum,
                             const float* __restrict__ V,
                             float* __restrict__ out, int E) {
    const long long t = (long long)blockIdx.x * blockDim.x + threadIdx.x;
    if (t >= (long long)E * (EMB / 4)) return;
    const int e = (int)(t >> 4);       // /16 chunks
    const int c = (int)(t & 15);
    const int u = users[e];
    const int it = items[e];
    const float alpha = ex[e] / segsum[u];
    const float4 v = ((const float4*)(V + (size_t)it * EMB))[c];
    float* dst = out + (size_t)u * EMB + c * 4;
    atomicAdd(dst + 0, v.x * alpha);
    atomicAdd(dst + 1, v.y * alpha);
    atomicAdd(dst + 2, v.z * alpha);
    atomicAdd(dst + 3, v.w * alpha);
}

// ---------------------------------------------------------------------------
// Kernel 6: scatter_mean finalize: out /= max(cnt, 1)
// ---------------------------------------------------------------------------
__global__ void finalize_mean(float* __restrict__ out,
                              const float* __restrict__ cnt, int NU) {
    const int t = blockIdx.x * blockDim.x + threadIdx.x;
    if (t >= NU * EMB) return;
    out[t] = out[t] / fmaxf(cnt[t >> 6], 1.0f);
}

// ---------------------------------------------------------------------------
extern "C" void kernel_launch(void* const* d_in, const int* in_sizes, int n_in,
                              void* d_out, int out_size, void* d_ws, size_t ws_size,
                              hipStream_t stream) {
    const int*   inter = (const int*)d_in[0];    // [2, E] stacked
    const float* ew    = (const float*)d_in[1];  // [E]
    const float* uemb  = (const float*)d_in[2];  // [NU, 64]
    const float* iemb  = (const float*)d_in[3];  // [NI, 64]
    const float* Wq    = (const float*)d_in[4];  // [64, 64]
    const float* Wk    = (const float*)d_in[5];
    const float* Wv    = (const float*)d_in[6];
    float* out = (float*)d_out;

    const int E  = in_sizes[1];
    const int NU = in_sizes[2] / EMB;
    const int NI = in_sizes[3] / EMB;

    const int* users = inter;
    const int* items = inter + E;

    // workspace layout (all float-sized, 16B-aligned chunks)
    float* Q       = (float*)d_ws;               // NU*64
    float* Kp      = Q + (size_t)NU * EMB;       // NI*64
    float* Vp      = Kp + (size_t)NI * EMB;      // NI*64
    float* scores  = Vp + (size_t)NI * EMB;      // E
    float* ex      = scores + E;                 // E
    float* segsum  = ex + E;                     // NU
    float* cnt     = segsum + NU;                // NU
    unsigned* segmax = (unsigned*)(cnt + NU);    // NU

    // 1) projections via fp32 WMMA (exact): Q = U@Wq.T, K = I@Wk.T, V = I@Wv.T
    {
        const int threads = 256;                  // 8 waves/block
        const int wavesU = (NU / 16) * 4;
        const int wavesI = (NI / 16) * 4;
        hipLaunchKernelGGL(proj_wmma_f32, dim3((wavesU + 7) / 8), dim3(threads), 0, stream,
                           uemb, Wq, Q, NU);
        hipLaunchKernelGGL(proj_wmma_f32, dim3((wavesI + 7) / 8), dim3(threads), 0, stream,
                           iemb, Wk, Kp, NI);
        hipLaunchKernelGGL(proj_wmma_f32, dim3((wavesI + 7) / 8), dim3(threads), 0, stream,
                           iemb, Wv, Vp, NI);
    }

    // 2) zero accumulators + output
    {
        const int n = NU * EMB;
        hipLaunchKernelGGL(init_zero, dim3((n + 255) / 256), dim3(256), 0, stream,
                           out, segsum, cnt, segmax, NU);
    }

    // 3) scores + segment max
    hipLaunchKernelGGL(edge_scores, dim3((E + 255) / 256), dim3(256), 0, stream,
                       users, items, ew, Q, Kp, scores, segmax, E);

    // 4) exp + segment sum + count
    hipLaunchKernelGGL(edge_exp, dim3((E + 255) / 256), dim3(256), 0, stream,
                       users, scores, segmax, ex, segsum, cnt, E);

    // 5) scatter alpha * value
    {
        const long long tot = (long long)E * (EMB / 4);
        hipLaunchKernelGGL(edge_scatter, dim3((unsigned)((tot + 255) / 256)), dim3(256), 0, stream,
                           users, items, ex, segsum, Vp, out, E);
    }

    // 6) mean
    {
        const int n = NU * EMB;
        hipLaunchKernelGGL(finalize_mean, dim3((n + 255) / 256), dim3(256), 0, stream,
                           out, cnt, NU);
    }
}